// Lfm2ShortConv_82575041233068
// MI455X (gfx1250) — compile-verified
//
#include <hip/hip_runtime.h>

// Problem constants (from reference)
#define B_SZ   256
#define H_SZ   2048
#define POOL   8192

typedef float v2f __attribute__((ext_vector_type(2)));
typedef float v8f __attribute__((ext_vector_type(8)));

#define BM  16
#define BN  128
#define BK  32
#define LDT 36   // padded LDS row stride (floats); keeps float4 alignment, avoids bank conflicts

// ---------------------------------------------------------------------------
// C[M,N] = A[M,K] * Bw[N,K]^T   (A, Bw row-major with K contiguous == "x @ W.T")
// Block: 256 threads = 8 waves. Block tile: 16(M) x 128(N). Wave w owns the
// 16x16 sub-tile at N columns [tileN + 16w, +16). K staged in LDS, BK=32.
//
// 4 independent accumulator chains (k-step j -> chain j&3) so the scheduler
// can interleave WMMAs instead of serializing on a single D->C dependency.
//
// WMMA fragment layout (ISA 7.12.2, f32 16x16x4):
//   A: lane = M row (lane&15); VGPR0/1 = K {0,1} (lanes 0-15) or {2,3} (lanes 16-31)
//   B: lane = N col (lane&15); VGPR0/1 = K rows, split by lane half like A
//   C/D: VGPR r -> M = r + 8*(lane>>4), N = lane&15
// ---------------------------------------------------------------------------
__global__ __launch_bounds__(256) void gemm_nt_wmma_f32(
    const float* __restrict__ A, const float* __restrict__ Bw,
    float* __restrict__ C, int N, int K)
{
    __shared__ __align__(16) float As[BM * LDT];
    __shared__ __align__(16) float Bs[BN * LDT];

    const int tid    = threadIdx.x;
    const int lane   = tid & 31;
    const int wave   = tid >> 5;       // 0..7
    const int lane16 = lane & 15;
    const int laneHi = lane >> 4;      // 0 or 1
    const int tileM  = blockIdx.y * BM;
    const int tileN  = blockIdx.x * BN;

    v8f acc[4];
    acc[0] = (v8f){};
    acc[1] = (v8f){};
    acc[2] = (v8f){};
    acc[3] = (v8f){};

    // Per-thread staging coordinates
    const int ar = tid >> 4;            // A row   0..15
    const int ac = (tid & 15) * 2;      // A col   0..30
    const int bq = (tid & 7) * 4;       // B K-off 0,4,...,28
    const int br = tid >> 3;            // B row   0..31 (x4 rows, stride 32)

    for (int kt = 0; kt < K; kt += BK) {
        // ---- stage A tile: 16 rows x 32 cols, one float2 per thread
        {
            const float2 v = *(const float2*)&A[(size_t)(tileM + ar) * K + kt + ac];
            *(float2*)&As[ar * LDT + ac] = v;
        }
        // ---- stage B tile: 128 rows (N) x 32 cols (K), four float4 per thread
        #pragma unroll
        for (int i = 0; i < 4; ++i) {
            const int r = br + i * 32;     // 0..127
            const float4 v = *(const float4*)&Bw[(size_t)(tileN + r) * K + kt + bq];
            *(float4*)&Bs[r * LDT + bq] = v;
        }
        // ---- prefetch next K-tile of the weight stream into L2 (global_prefetch_b8)
        if (kt + BK < K) {
            #pragma unroll
            for (int i = 0; i < 4; ++i) {
                const int r = br + i * 32;
                __builtin_prefetch(&Bw[(size_t)(tileN + r) * K + kt + BK + bq], 0, 1);
            }
            __builtin_prefetch(&A[(size_t)(tileM + ar) * K + kt + BK + ac], 0, 1);
        }
        __syncthreads();

        // ---- 8 x v_wmma_f32_16x16x4_f32, round-robin over 4 accumulator chains
        #pragma unroll
        for (int j = 0; j < BK / 4; ++j) {
            const int ko = j * 4 + laneHi * 2;
            v2f a = *(const v2f*)&As[lane16 * LDT + ko];
            v2f b = *(const v2f*)&Bs[(wave * 16 + lane16) * LDT + ko];
            acc[j & 3] = __builtin_amdgcn_wmma_f32_16x16x4_f32(
                /*neg_a=*/false, a, /*neg_b=*/false, b,
                /*c_mod=*/(short)0, acc[j & 3],
                /*reuse_a=*/false, /*reuse_b=*/false);
        }
        __syncthreads();
    }

    const v8f sum = (acc[0] + acc[1]) + (acc[2] + acc[3]);

    // ---- writeback: VGPR r -> row (tileM + 8*laneHi + r), col (tileN + 16*wave + lane16)
    const int n     = tileN + wave * 16 + lane16;
    const int mBase = tileM + laneHi * 8;
    #pragma unroll
    for (int r = 0; r < 8; ++r) {
        C[(size_t)(mBase + r) * N + n] = sum[r];
    }
}

// ---------------------------------------------------------------------------
// Gating + depthwise K=3 conv + conv-state shift/scatter.
// proj layout: [b, 0:H]=B_gate, [b, H:2H]=C_gate, [b, 2H:3H]=x
// y[b,h] = C_gate * (c0*w0 + c1*w1 + (B_gate*x)*w2)
// state row idx[b]: [h,0] <- c1, [h,1] <- B_gate*x   (full-pool copy done
// earlier in stream order by hipMemcpyAsync; indices are a permutation)
// ---------------------------------------------------------------------------
__global__ __launch_bounds__(256) void gate_conv_scatter(
    const float* __restrict__ proj, const float* __restrict__ conv_state,
    const int* __restrict__ req_pool_indices, const float* __restrict__ conv_w,
    float* __restrict__ y, float* __restrict__ out_state)
{
    const int b   = blockIdx.x;
    const int row = req_pool_indices[b];
    const float* pb = proj + (size_t)b * (3 * H_SZ);

    for (int h = threadIdx.x; h < H_SZ; h += blockDim.x) {
        const float Bg = pb[h];
        const float Cg = pb[H_SZ + h];
        const float xx = pb[2 * H_SZ + h];
        const float Bx = Bg * xx;

        const float2 cur = *(const float2*)&conv_state[(size_t)row * (2 * H_SZ) + 2 * h];
        const float w0 = conv_w[h * 3 + 0];
        const float w1 = conv_w[h * 3 + 1];
        const float w2 = conv_w[h * 3 + 2];

        const float co = fmaf(cur.x, w0, fmaf(cur.y, w1, Bx * w2));
        y[(size_t)b * H_SZ + h] = Cg * co;

        float2 ns;
        ns.x = cur.y;
        ns.y = Bx;
        *(float2*)&out_state[(size_t)row * (2 * H_SZ) + 2 * h] = ns;
    }
}

// ---------------------------------------------------------------------------
extern "C" void kernel_launch(void* const* d_in, const int* in_sizes, int n_in,
                              void* d_out, int out_size, void* d_ws, size_t ws_size,
                              hipStream_t stream) {
    const float* hs         = (const float*)d_in[0];   // (256, 2048)
    const float* conv_state = (const float*)d_in[1];   // (8192, 2048, 2)
    const int*   idx        = (const int*)d_in[2];     // (256,)
    const float* W_in       = (const float*)d_in[3];   // (6144, 2048)
    const float* W_out      = (const float*)d_in[4];   // (2048, 2048)
    const float* conv_w     = (const float*)d_in[5];   // (2048, 3)

    float* out       = (float*)d_out;                        // (256, 2048)
    float* out_state = out + (size_t)B_SZ * H_SZ;            // (8192, 2048, 2)

    float* proj = (float*)d_ws;                              // (256, 6144)
    float* y    = proj + (size_t)B_SZ * 3 * H_SZ;            // (256, 2048)

    // 1) Bulk-copy pool state into the output (268 MB of HBM traffic — the
    //    roofline floor for this op). Stream-ordered before the row scatter.
    hipMemcpyAsync(out_state, conv_state,
                   (size_t)POOL * H_SZ * 2 * sizeof(float),
                   hipMemcpyDeviceToDevice, stream);

    // 2) proj = hs @ W_in^T   (M=256, N=6144, K=2048)
    gemm_nt_wmma_f32<<<dim3(3 * H_SZ / BN, B_SZ / BM), 256, 0, stream>>>(
        hs, W_in, proj, 3 * H_SZ, H_SZ);

    // 3) gating + depthwise conv + state-row scatter
    gate_conv_scatter<<<B_SZ, 256, 0, stream>>>(
        proj, conv_state, idx, conv_w, y, out_state);

    // 4) out = y @ W_out^T    (M=256, N=2048, K=2048)
    gemm_nt_wmma_f32<<<dim3(H_SZ / BN, B_SZ / BM), 256, 0, stream>>>(
        y, W_out, out, H_SZ, H_SZ);
}